// EMHA_41188736369439
// MI455X (gfx1250) — compile-verified
//
#include <hip/hip_runtime.h>
#include <cstdint>
#include <cstddef>

// fp32 WMMA fragment types for V_WMMA_F32_16X16X4_F32:
//  A (16x4) = 2 VGPRs/lane, B (4x16) = 2 VGPRs/lane, C/D (16x16) = 8 VGPRs/lane
typedef float v2f __attribute__((ext_vector_type(2)));
typedef float v8f __attribute__((ext_vector_type(8)));

#define EMBED 1024
#define NHEAD 8
#define SFAC 4
#define HD 64
#define BATCH 4
#define SEQ 4096
#define NS (SEQ / SFAC)            // 1024 rows/cols per attention group
#define ROWS_PER_B (SEQ * NHEAD)   // 32768 rows of the (rows x 64) h/q/k/v view

static __device__ __forceinline__ v8f wmma4(v2f a, v2f b, v8f c) {
  // 8 args: (neg_a, A, neg_b, B, c_mod, C, reuse_a, reuse_b)
  return __builtin_amdgcn_wmma_f32_16x16x4_f32(false, a, false, b, (short)0, c,
                                               false, false);
}

static __device__ __forceinline__ v2f ld2g(const float* p) {
  return *reinterpret_cast<const v2f*>(p);
}
static __device__ __forceinline__ v2f ld2s(const float* p) {
  return *reinterpret_cast<const v2f*>(p);
}

// ---------------------------------------------------------------------------
// C[b] (Mo x 4096) = W (Mo x K) @ X[b] (K x 4096) + bias   (conv1x1 stages)
// block = 128 threads (4 waves). Each wave owns a 64-row x 16-col output
// panel (4 stacked 16x16 WMMA tiles) so one B fragment feeds 4 WMMAs per
// K-step: 4x less re-read traffic on the large X operand (L2-resident).
// ---------------------------------------------------------------------------
__global__ void gemm_bias_f32(const float* __restrict__ W,
                              const float* __restrict__ X,
                              const float* __restrict__ bias,
                              float* __restrict__ C, int Mo, int K) {
  const int b    = blockIdx.z;
  const int lane = threadIdx.x & 31;
  const int wave = threadIdx.x >> 5;
  const int lo   = lane & 15;
  const int hi   = lane >> 4;

  const float* Xb = X + (size_t)b * K * SEQ;
  float*       Cb = C + (size_t)b * Mo * SEQ;

  const int rowBase = blockIdx.y * 64;
  const int col     = blockIdx.x * 64 + wave * 16 + lo;

  // A: lane lo supplies row (rowBase + t*16 + lo), K-pair 2*hi
  const float* wrow0 = W + (size_t)(rowBase + lo) * K + 2 * hi;
  // B: lane lo supplies column `col`, rows kk+2*hi / kk+2*hi+1
  const float* xcol  = Xb + (size_t)(2 * hi) * SEQ + col;

  v8f acc[4] = {};
  for (int kk = 0; kk < K; kk += 4) {
    v2f bb;
    bb.x = xcol[(size_t)kk * SEQ];
    bb.y = xcol[(size_t)kk * SEQ + SEQ];
    // stream-prefetch the X panel ahead (lowers to global_prefetch_b8)
    if (kk + 256 < K)
      __builtin_prefetch(xcol + (size_t)(kk + 256) * SEQ, 0, 1);
#pragma unroll
    for (int t = 0; t < 4; ++t) {
      v2f a = ld2g(wrow0 + (size_t)t * 16 * K + kk);
      acc[t] = wmma4(a, bb, acc[t]);
    }
  }
#pragma unroll
  for (int t = 0; t < 4; ++t)
#pragma unroll
    for (int v = 0; v < 8; ++v) {
      const int row = rowBase + t * 16 + v + 8 * hi;  // C/D: VGPR v -> rows v / v+8
      Cb[(size_t)row * SEQ + col] = acc[t][v] + bias[row];
    }
}

// ---------------------------------------------------------------------------
// q/k/v = h @ W{q,k,v}^T ; h is the xr buffer viewed flat as (B*32768, 64)
// One A fragment feeds three accumulators per K step.
// ---------------------------------------------------------------------------
__global__ void qkv_proj(const float* __restrict__ H,
                         const float* __restrict__ Wq,
                         const float* __restrict__ Wk,
                         const float* __restrict__ Wv,
                         float* __restrict__ Q,
                         float* __restrict__ Kd,
                         float* __restrict__ V) {
  const int lane = threadIdx.x & 31;
  const int wave = threadIdx.x >> 5;
  const int lo = lane & 15, hi = lane >> 4;

  const int rowBase = blockIdx.x * 16;
  const int o       = wave * 16 + lo;                  // output channel (0..63)

  const float* hrow = H + (size_t)(rowBase + lo) * HD + 2 * hi;
  // B[e][o] = W[o][e] -> each lane reads 2 consecutive floats of row o
  const float* wqr = Wq + (size_t)o * HD + 2 * hi;
  const float* wkr = Wk + (size_t)o * HD + 2 * hi;
  const float* wvr = Wv + (size_t)o * HD + 2 * hi;

  v8f aq = {}, ak = {}, av = {};
#pragma unroll
  for (int kk = 0; kk < HD; kk += 4) {
    v2f a = ld2g(hrow + kk);
    aq = wmma4(a, ld2g(wqr + kk), aq);
    ak = wmma4(a, ld2g(wkr + kk), ak);
    av = wmma4(a, ld2g(wvr + kk), av);
  }
#pragma unroll
  for (int v = 0; v < 8; ++v) {
    const int row = rowBase + v + 8 * hi;
    Q [(size_t)row * HD + o] = aq[v];
    Kd[(size_t)row * HD + o] = ak[v];
    V [(size_t)row * HD + o] = av[v];
  }
}

// ---------------------------------------------------------------------------
// Flash attention per group g=(b,m,s): rows of q/k/v at R = i*32 + m*4 + s.
// 4 waves/block; each wave owns one 16-row i-tile, streams 64 j-tiles of 16.
// Online softmax (scale 1/32), P transposed via 1KB LDS into A-frag layout.
// ---------------------------------------------------------------------------
__global__ void flash_attn(const float* __restrict__ Q,
                           const float* __restrict__ K,
                           const float* __restrict__ V,
                           float* __restrict__ O) {
  const int lane = threadIdx.x & 31;
  const int wave = threadIdx.x >> 5;
  const int lo = lane & 15, hi = lane >> 4;

  const int itile = blockIdx.x * 4 + wave;             // 0..63
  const int g = blockIdx.y;                            // 0..127
  const int b = g >> 5, m = (g >> 2) & 7, s = g & 3;
  const int boff = b * ROWS_PER_B;
  const int rco  = m * SFAC + s;

  __shared__ float Plds[4][16 * 16];
  float* pl = Plds[wave];

  const float scale = 0.03125f;  // 1/sqrt(EMBED)

  const float* qrow =
      Q + (size_t)(boff + (itile * 16 + lo) * 32 + rco) * HD + 2 * hi;

  // lanes 0-15 track tile-rows 0..7 (per VGPR v), lanes 16-31 rows 8..15
  float mrow[8], lrow[8];
#pragma unroll
  for (int v = 0; v < 8; ++v) { mrow[v] = -3.0e38f; lrow[v] = 0.0f; }

  v8f oacc[4] = {};

  for (int jt = 0; jt < NS / 16; ++jt) {
    // ---- E = Q @ K^T (16x16), K streamed as B[e][j] = k[jrow][e] ----
    const float* krow =
        K + (size_t)(boff + (jt * 16 + lo) * 32 + rco) * HD + 2 * hi;
    v8f e = {};
#pragma unroll
    for (int kk = 0; kk < HD; kk += 4)
      e = wmma4(ld2g(qrow + kk), ld2g(krow + kk), e);

    // ---- online softmax: row max / exp / row sum (halves stay disjoint) ----
    float alpha[8];
#pragma unroll
    for (int v = 0; v < 8; ++v) {
      float xv = e[v] * scale;
      float rmax = xv;
      rmax = fmaxf(rmax, __shfl_xor(rmax, 1, 32));
      rmax = fmaxf(rmax, __shfl_xor(rmax, 2, 32));
      rmax = fmaxf(rmax, __shfl_xor(rmax, 4, 32));
      rmax = fmaxf(rmax, __shfl_xor(rmax, 8, 32));
      float mnew = fmaxf(mrow[v], rmax);
      float p = __expf(xv - mnew);
      float rsum = p;
      rsum += __shfl_xor(rsum, 1, 32);
      rsum += __shfl_xor(rsum, 2, 32);
      rsum += __shfl_xor(rsum, 4, 32);
      rsum += __shfl_xor(rsum, 8, 32);
      float a = __expf(mrow[v] - mnew);
      lrow[v] = lrow[v] * a + rsum;
      mrow[v] = mnew;
      alpha[v] = a;
      pl[(v + 8 * hi) * 16 + lo] = p;   // P[row][col] into LDS
    }
#pragma unroll
    for (int c = 0; c < 4; ++c)
#pragma unroll
      for (int v = 0; v < 8; ++v) oacc[c][v] *= alpha[v];

    __builtin_amdgcn_wave_barrier();    // order LDS store -> load (in-wave)

    // ---- O += P @ V : A from LDS (row-major P), B from v rows ----
#pragma unroll
    for (int kk = 0; kk < 16; kk += 4) {
      v2f a = ld2s(&pl[lo * 16 + kk + 2 * hi]);
      const int j0 = jt * 16 + kk + 2 * hi;
      const float* vr0 = V + (size_t)(boff + j0 * 32 + rco) * HD;
      const float* vr1 = V + (size_t)(boff + (j0 + 1) * 32 + rco) * HD;
#pragma unroll
      for (int c = 0; c < 4; ++c) {
        v2f bb;
        bb.x = vr0[c * 16 + lo];
        bb.y = vr1[c * 16 + lo];
        oacc[c] = wmma4(a, bb, oacc[c]);
      }
    }
    __builtin_amdgcn_wave_barrier();
  }

  // ---- normalize and scatter O rows back at R = i*32 + m*4 + s ----
#pragma unroll
  for (int v = 0; v < 8; ++v) {
    const float inv = 1.0f / lrow[v];
    const int grow = boff + (itile * 16 + v + 8 * hi) * 32 + rco;
#pragma unroll
    for (int c = 0; c < 4; ++c)
      O[(size_t)grow * HD + c * 16 + lo] = oacc[c][v] * inv;
  }
}

// ---------------------------------------------------------------------------
extern "C" void kernel_launch(void* const* d_in, const int* in_sizes, int n_in,
                              void* d_out, int out_size, void* d_ws, size_t ws_size,
                              hipStream_t stream) {
  const float* x  = (const float*)d_in[0];
  const float* Wq = (const float*)d_in[1];
  const float* Wk = (const float*)d_in[2];
  const float* Wv = (const float*)d_in[3];
  const float* Wr = (const float*)d_in[4];
  const float* br = (const float*)d_in[5];
  const float* We = (const float*)d_in[6];
  const float* be = (const float*)d_in[7];
  float* out = (float*)d_out;

  // Workspace layout (fp32): xr (reused as O) | q | k | v  -> 128 MB total
  const size_t perB = (size_t)ROWS_PER_B * HD;      // 2,097,152 floats
  float* xrO = (float*)d_ws;
  float* q   = xrO + (size_t)BATCH * perB;
  float* k   = q   + (size_t)BATCH * perB;
  float* v   = k   + (size_t)BATCH * perB;

  // 1) reduction conv1x1: xr = Wr @ x + br          (512 x 4096 x 1024, B=4)
  gemm_bias_f32<<<dim3(SEQ / 64, 512 / 64, BATCH), 128, 0, stream>>>(
      Wr, x, br, xrO, 512, EMBED);

  // 2) per-head projections: q/k/v = h @ W^T        (131072 x 64 x 64)
  qkv_proj<<<dim3(BATCH * ROWS_PER_B / 16), 128, 0, stream>>>(
      xrO, Wq, Wk, Wv, q, k, v);

  // 3) flash attention per (b,m,s) group; writes O over the xr buffer
  flash_attn<<<dim3(NS / 16 / 4, BATCH * NHEAD * SFAC), 128, 0, stream>>>(
      q, k, v, xrO);

  // 4) expansion conv1x1: out = We @ O + be         (1024 x 4096 x 512, B=4)
  gemm_bias_f32<<<dim3(SEQ / 64, EMBED / 64, BATCH), 128, 0, stream>>>(
      We, xrO, be, out, EMBED, 512);
}